// PoseGraph_58531814310215
// MI455X (gfx1250) — compile-verified
//
#include <hip/hip_runtime.h>

#define EPSF 1e-6f
#define BLK 256

typedef int v2i __attribute__((ext_vector_type(2)));

// ---- CDNA5 async LDS staging (gated: falls back cleanly if toolchain lacks it)
#if defined(__has_builtin)
#  if __has_builtin(__builtin_amdgcn_global_load_async_to_lds_b32) && \
      __has_builtin(__builtin_amdgcn_s_wait_asynccnt)
#    define ASYNC_LOAD 1
#  endif
#  if __has_builtin(__builtin_amdgcn_global_store_async_from_lds_b32) && \
      __has_builtin(__builtin_amdgcn_s_wait_asynccnt)
#    define ASYNC_STORE 1
#  endif
#endif

typedef __attribute__((address_space(1))) int gint;  // global b32 word
typedef __attribute__((address_space(3))) int lint;  // LDS b32 word

// ---------------- SE(3) math (faithful port of the reference) ----------------

__device__ __forceinline__ void qrot(const float* __restrict__ q,
                                     const float* __restrict__ v,
                                     float* __restrict__ o) {
    float tx = 2.0f * (q[1] * v[2] - q[2] * v[1]);
    float ty = 2.0f * (q[2] * v[0] - q[0] * v[2]);
    float tz = 2.0f * (q[0] * v[1] - q[1] * v[0]);
    o[0] = v[0] + q[3] * tx + (q[1] * tz - q[2] * ty);
    o[1] = v[1] + q[3] * ty + (q[2] * tx - q[0] * tz);
    o[2] = v[2] + q[3] * tz + (q[0] * ty - q[1] * tx);
}

__device__ __forceinline__ void qmul(const float* __restrict__ q,
                                     const float* __restrict__ r,
                                     float* __restrict__ o) {
    o[0] = q[3] * r[0] + q[0] * r[3] + q[1] * r[2] - q[2] * r[1];
    o[1] = q[3] * r[1] - q[0] * r[2] + q[1] * r[3] + q[2] * r[0];
    o[2] = q[3] * r[2] + q[0] * r[1] - q[1] * r[0] + q[2] * r[3];
    o[3] = q[3] * r[3] - q[0] * r[0] - q[1] * r[1] - q[2] * r[2];
}

__device__ __forceinline__ void se3_inv(const float* __restrict__ T,
                                        float* __restrict__ o) {
    float qc[4] = {-T[3], -T[4], -T[5], T[6]};
    float ti[3];
    qrot(qc, T, ti);
    o[0] = -ti[0]; o[1] = -ti[1]; o[2] = -ti[2];
    o[3] = qc[0];  o[4] = qc[1];  o[5] = qc[2];  o[6] = qc[3];
}

__device__ __forceinline__ void se3_mul(const float* __restrict__ A,
                                        const float* __restrict__ B,
                                        float* __restrict__ o) {
    float r[3];
    qrot(A + 3, B, r);
    o[0] = A[0] + r[0];
    o[1] = A[1] + r[1];
    o[2] = A[2] + r[2];
    qmul(A + 3, B + 3, o + 3);
}

// res[0..5] = se3_log( (T^-1 * n2) * n1^-1 )
__device__ __forceinline__ void se3_error_log(const float* __restrict__ T,
                                              const float* __restrict__ n1,
                                              const float* __restrict__ n2,
                                              float* __restrict__ res) {
    float Pinv[7], N1inv[7], M[7], E[7];
    se3_inv(T, Pinv);
    se3_inv(n1, N1inv);
    se3_mul(Pinv, n2, M);
    se3_mul(M, N1inv, E);

    // so3_log
    float vx = E[3], vy = E[4], vz = E[5], w = E[6];
    float nn = sqrtf(vx * vx + vy * vy + vz * vz);
    float scale;
    if (nn > EPSF) {
        scale = 2.0f * atan2f(nn, w) / nn;
    } else {
        scale = 2.0f / ((fabsf(w) > EPSF) ? w : 1.0f);
    }
    float px = vx * scale, py = vy * scale, pz = vz * scale;

    float theta = sqrtf(px * px + py * py + pz * pz);
    float c;
    if (theta < EPSF) {
        c = 1.0f / 12.0f;
    } else {
        c = 1.0f / (theta * theta) -
            (1.0f + cosf(theta)) / (2.0f * theta * sinf(theta));
    }

    float tx = E[0], ty = E[1], tz = E[2];
    float pxt0 = py * tz - pz * ty;
    float pxt1 = pz * tx - px * tz;
    float pxt2 = px * ty - py * tx;
    float ppxt0 = py * pxt2 - pz * pxt1;
    float ppxt1 = pz * pxt0 - px * pxt2;
    float ppxt2 = px * pxt1 - py * pxt0;

    res[0] = tx - 0.5f * pxt0 + c * ppxt0;
    res[1] = ty - 0.5f * pxt1 + c * ppxt1;
    res[2] = tz - 0.5f * pxt2 + c * ppxt2;
    res[3] = px;
    res[4] = py;
    res[5] = pz;
}

// ------------------------------- kernel --------------------------------------

__global__ __launch_bounds__(BLK)
void PoseGraph_kernel(const int* __restrict__ edges,
                      const float* __restrict__ poses,
                      const float* __restrict__ nodes,
                      float* __restrict__ out,
                      int nEdges) {
    __shared__ float smem[7 * BLK];

    const int tid = threadIdx.x;
    const int blkBase = blockIdx.x * BLK;   // fits in int (grid = 12500)

    if (blkBase + BLK <= nEdges) {
        // ============ FAST PATH: whole block in range, no per-lane guards ====

        // 1) Kick off pose staging (async global->LDS when available).
        const float* pSlab = poses + (size_t)blkBase * 7;
#if ASYNC_LOAD
        gint* g = (gint*)pSlab;
        lint* l = (lint*)smem;
#pragma unroll
        for (int k = 0; k < 7; ++k) {
            int off = k * BLK + tid;
            __builtin_amdgcn_global_load_async_to_lds_b32(g + off, l + off, 0, 0);
        }
#else
#pragma unroll
        for (int k = 0; k < 7; ++k) {
            int off = k * BLK + tid;
            smem[off] = __builtin_nontemporal_load(&pSlab[off]);
        }
#endif

        // 2) Edge indices + node gathers in flight while staging proceeds.
        v2i ee = __builtin_nontemporal_load((const v2i*)edges + (blkBase + tid));
        const float* __restrict__ p1 = nodes + ee.x * 7;   // table <= 2.8 MB: int offsets
        const float* __restrict__ p2 = nodes + ee.y * 7;
        float n1[7], n2[7];
#pragma unroll
        for (int k = 0; k < 7; ++k) { n1[k] = p1[k]; n2[k] = p2[k]; }

        // 3) Wait for staged poses, then read own record from LDS.
#if ASYNC_LOAD
        __builtin_amdgcn_s_wait_asynccnt(0);
#endif
        __syncthreads();
        float T[7];
#pragma unroll
        for (int k = 0; k < 7; ++k) T[k] = smem[tid * 7 + k];

        // 4) Math.
        float res[6];
        se3_error_log(T, n1, n2, res);

        // 5) Stage result to LDS, write out coalesced (async when available).
        __syncthreads();
#pragma unroll
        for (int k = 0; k < 6; ++k) smem[tid * 6 + k] = res[k];
        __syncthreads();

        float* oSlab = out + (size_t)blkBase * 6;
#if ASYNC_STORE
        gint* go = (gint*)oSlab;
        lint* lo = (lint*)smem;
#pragma unroll
        for (int k = 0; k < 6; ++k) {
            int off = k * BLK + tid;
            __builtin_amdgcn_global_store_async_from_lds_b32(go + off, lo + off, 0, 0);
        }
        // S_ENDPGM performs an implicit wait-idle; no explicit asynccnt wait needed.
#else
#pragma unroll
        for (int k = 0; k < 6; ++k) {
            int off = k * BLK + tid;
            __builtin_nontemporal_store(smem[k * BLK + tid], &oSlab[off]);
        }
#endif
    } else {
        // ============ TAIL PATH (never taken for N_EDGES % 256 == 0) =========
        const int e = blkBase + tid;
        if (e < nEdges) {
            float T[7];
#pragma unroll
            for (int k = 0; k < 7; ++k) T[k] = poses[(size_t)e * 7 + k];
            int i1 = edges[2 * (size_t)e];
            int i2 = edges[2 * (size_t)e + 1];
            const float* p1 = nodes + i1 * 7;
            const float* p2 = nodes + i2 * 7;
            float n1[7], n2[7];
#pragma unroll
            for (int k = 0; k < 7; ++k) { n1[k] = p1[k]; n2[k] = p2[k]; }
            float res[6];
            se3_error_log(T, n1, n2, res);
#pragma unroll
            for (int k = 0; k < 6; ++k) out[(size_t)e * 6 + k] = res[k];
        }
    }
}

extern "C" void kernel_launch(void* const* d_in, const int* in_sizes, int n_in,
                              void* d_out, int out_size, void* d_ws, size_t ws_size,
                              hipStream_t stream) {
    const int*   edges = (const int*)d_in[0];    // (N_EDGES, 2) int32
    const float* poses = (const float*)d_in[1];  // (N_EDGES, 7) f32
    const float* nodes = (const float*)d_in[2];  // (N_NODES, 7) f32
    float*       out   = (float*)d_out;          // (N_EDGES, 6) f32

    const int nEdges = in_sizes[1] / 7;
    const int grid = (nEdges + BLK - 1) / BLK;

    PoseGraph_kernel<<<grid, BLK, 0, stream>>>(edges, poses, nodes, out, nEdges);
}